// TestingHMM_39032662786338
// MI455X (gfx1250) — compile-verified
//
#include <hip/hip_runtime.h>

// ---------------------------------------------------------------------------
// HMM forward pass for MI455X (gfx1250, wave32).
//   B=256 batch, T=64 steps, K=512 states.
//   Per step: trans[i,j,k] = softmax_k(post[i,j]*coeff[j,k] + bias[j,k])
//             pbe[i,k]     = sum_j post[i,j] * trans[i,j,k]
//             post'        = rownorm(pbe * evidence(x_t))
// TRANS-bound (4.2G exp). Hot loop runs in log2 domain:
//   e = v_exp_f32( fma(pj*log2e, coeff, bias*log2e) )   -> 3 VALU : 1 TRANS
// bias*log2e precomputed once into ws (1 MB, L2-resident, reused 256x63).
// All softmax reciprocals use v_rcp_f32 (no IEEE div chains).
// Cross-wave pbe-partial reduction runs on the matrix pipe via
// V_WMMA_F32_16X16X4_F32 (A = ones => layout-invariant column sums).
// ---------------------------------------------------------------------------

typedef __attribute__((ext_vector_type(2))) float v2f;
typedef __attribute__((ext_vector_type(8))) float v8f;

#define K_  512
#define NW  16      // waves per block (512 threads, wave32)

#define LOG2E 1.4426950408889634f

__device__ __forceinline__ float waveSum(float v) {
#pragma unroll
  for (int m = 16; m >= 1; m >>= 1) v += __shfl_xor(v, m, 32);
  return v;
}

__device__ __forceinline__ float fastRcp(float v) {
  return __builtin_amdgcn_rcpf(v);          // v_rcp_f32, 1 TRANS op
}
__device__ __forceinline__ float fastExp2(float v) {
  return __builtin_amdgcn_exp2f(v);         // v_exp_f32, 1 TRANS op
}

// bias2 = bias * log2(e), staged once into workspace
__global__ __launch_bounds__(256) void scale_bias(const float* __restrict__ b,
                                                  float* __restrict__ b2, int n4) {
  const int idx = blockIdx.x * blockDim.x + threadIdx.x;
  if (idx < n4) {
    float4 v = ((const float4*)b)[idx];
    v.x *= LOG2E; v.y *= LOG2E; v.z *= LOG2E; v.w *= LOG2E;
    ((float4*)b2)[idx] = v;
  }
}

// post0[i,k] = normalize_k( exp(prior_w[k]) * ((1-x0)*em0[k] + x0*em1[k]) )
// (softmax normalizers cancel under the row normalization)
__global__ __launch_bounds__(K_) void hmm_init(const float* __restrict__ x,
                                               const float* __restrict__ prior_w,
                                               const float* __restrict__ emw,
                                               float* __restrict__ post0, int T) {
  __shared__ float s_red[NW];
  const int i = blockIdx.x;
  const int k = threadIdx.x;
  const int lane = k & 31, wave = k >> 5;

  const float x0 = x[(size_t)i * T];
  const float pe = fastExp2(prior_w[k] * LOG2E);
  const float w0 = emw[2 * k], w1 = emw[2 * k + 1];
  const float e0 = fastRcp(1.f + fastExp2((w1 - w0) * LOG2E));  // sigmoid
  const float ev = fmaf(x0, 1.f - 2.f * e0, e0);                // (1-x)*e0+x*e1
  const float u  = pe * ev;

  float s = waveSum(u);
  if (lane == 0) s_red[wave] = s;
  __syncthreads();
  if (wave == 0) {
    float t2 = (lane < NW) ? s_red[lane] : 0.f;
    t2 = waveSum(t2);
    if (lane == 0) s_red[0] = t2;
  }
  __syncthreads();
  post0[(size_t)i * K_ + k] = u * fastRcp(s_red[0]);
}

// SCALED=1: biasP already holds bias*log2e. SCALED=0: scale in-loop.
template <int SCALED>
__global__ __launch_bounds__(K_) void hmm_step(const float* __restrict__ postIn,
                                               float* __restrict__ postOut,
                                               const float* __restrict__ coeff,
                                               const float* __restrict__ biasP,
                                               const float* __restrict__ emw,
                                               const float* __restrict__ x,
                                               int t, int T) {
  __shared__ float s_post[K_];
  __shared__ float s_part[NW][K_];
  __shared__ float s_red[NW];

  const int i = blockIdx.x;
  const int tid = threadIdx.x;
  const int lane = tid & 31, wave = tid >> 5;

  s_post[tid] = postIn[(size_t)i * K_ + tid];
  __syncthreads();

  // ---- transition + contraction over this wave's 32 j-rows -----------------
  float4 acc[4];
#pragma unroll
  for (int cc = 0; cc < 4; ++cc) acc[cc] = make_float4(0.f, 0.f, 0.f, 0.f);

  const int jlo = wave * 32;
#pragma unroll 2
  for (int j = jlo; j < jlo + 32; ++j) {
    const float pj  = s_post[j];                     // uniform LDS broadcast
    const float pjl = pj * LOG2E;
    const float4* c4 = (const float4*)(coeff + (size_t)j * K_);
    const float4* b4 = (const float4*)(biasP + (size_t)j * K_);
    float4 ec[4];
    float z = 0.f;
#pragma unroll
    for (int cc = 0; cc < 4; ++cc) {                 // cols cc*128 + lane*4 + e
      const float4 cv = c4[cc * 32 + lane];
      float4 bv = b4[cc * 32 + lane];
      if (!SCALED) { bv.x *= LOG2E; bv.y *= LOG2E; bv.z *= LOG2E; bv.w *= LOG2E; }
      float4 e;
      e.x = fastExp2(fmaf(pjl, cv.x, bv.x));
      e.y = fastExp2(fmaf(pjl, cv.y, bv.y));
      e.z = fastExp2(fmaf(pjl, cv.z, bv.z));
      e.w = fastExp2(fmaf(pjl, cv.w, bv.w));
      ec[cc] = e;
      z += (e.x + e.y) + (e.z + e.w);
    }
    z = waveSum(z);                                  // Z[i,j] over all k
    const float wj = pj * fastRcp(z);                // post/Z, v_rcp_f32
#pragma unroll
    for (int cc = 0; cc < 4; ++cc) {
      acc[cc].x = fmaf(wj, ec[cc].x, acc[cc].x);
      acc[cc].y = fmaf(wj, ec[cc].y, acc[cc].y);
      acc[cc].z = fmaf(wj, ec[cc].z, acc[cc].z);
      acc[cc].w = fmaf(wj, ec[cc].w, acc[cc].w);
    }
  }

  // per-wave partial pbe -> LDS, layout matches the column indexing above
  {
    float4* prow = (float4*)&s_part[wave][0];
#pragma unroll
    for (int cc = 0; cc < 4; ++cc) prow[cc * 32 + lane] = acc[cc];
  }
  __syncthreads();

  // ---- WMMA reduction of 16 partial rows (matrix pipe) ---------------------
  // D = ones(16x4) x P(4x16) chained 4x  => every row of D holds sum over the
  // 16 wave-partials for its 16 k-columns. A==ones makes the result invariant
  // to the exact A/B VGPR slot mapping (sum is permutation-invariant).
  v2f a; a.x = 1.f; a.y = 1.f;
  float res[2];
#pragma unroll
  for (int tt = 0; tt < 2; ++tt) {
    const int kt    = (wave * 2 + tt) * 16;
    const int col   = kt + (lane & 15);
    const int rbase = (lane >> 4) << 1;
    v8f d = {0.f, 0.f, 0.f, 0.f, 0.f, 0.f, 0.f, 0.f};
#pragma unroll
    for (int q = 0; q < 4; ++q) {
      v2f b;
      b.x = s_part[q * 4 + rbase][col];
      b.y = s_part[q * 4 + rbase + 1][col];
      d = __builtin_amdgcn_wmma_f32_16x16x4_f32(false, a, false, b,
                                                (short)0, d, false, false);
    }
    res[tt] = d[0];                                  // replicated over M rows
  }
  if (lane < 16) {
    s_post[(wave * 2 + 0) * 16 + lane] = res[0];     // s_post reused as pbe
    s_post[(wave * 2 + 1) * 16 + lane] = res[1];
  }
  __syncthreads();

  // ---- evidence + row normalization ---------------------------------------
  const float pbe = s_post[tid];
  const float w0 = emw[2 * tid], w1 = emw[2 * tid + 1];
  const float e0 = fastRcp(1.f + fastExp2((w1 - w0) * LOG2E));
  const float xt = x[(size_t)i * T + t];
  const float ev = fmaf(xt, 1.f - 2.f * e0, e0);
  const float u  = pbe * ev;

  float s = waveSum(u);
  if (lane == 0) s_red[wave] = s;
  __syncthreads();
  if (wave == 0) {
    float t2 = (lane < NW) ? s_red[lane] : 0.f;
    t2 = waveSum(t2);
    if (lane == 0) s_red[0] = t2;
  }
  __syncthreads();
  postOut[(size_t)i * K_ + tid] = u * fastRcp(s_red[0]);
}

extern "C" void kernel_launch(void* const* d_in, const int* in_sizes, int n_in,
                              void* d_out, int out_size, void* d_ws, size_t ws_size,
                              hipStream_t stream) {
  const float* x       = (const float*)d_in[0];   // [B,T]
  const float* prior_w = (const float*)d_in[1];   // [K]
  const float* emw     = (const float*)d_in[2];   // [K,2]
  const float* coeff   = (const float*)d_in[3];   // [K,K]
  const float* bias    = (const float*)d_in[4];   // [K,K]

  const int K = in_sizes[1];            // 512
  const int B = out_size / K;           // 256
  const int T = in_sizes[0] / B;        // 64

  float* out  = (float*)d_out;
  float* bufA = (float*)d_ws;           // B*K f32 ping buffer (512 KB)

  // scaled-bias staging area after the ping buffer, if ws is large enough
  const size_t pingBytes = (size_t)B * K * sizeof(float);
  const size_t needBytes = pingBytes + (size_t)K * K * sizeof(float);
  const bool useScaled = (ws_size >= needBytes);
  float* bias2 = (float*)((char*)d_ws + pingBytes);

  if (useScaled) {
    const int n4 = (K * K) / 4;
    hipLaunchKernelGGL(scale_bias, dim3((n4 + 255) / 256), dim3(256), 0, stream,
                       bias, bias2, n4);
  }

  float* p0 = (T > 1) ? bufA : out;
  hipLaunchKernelGGL(hmm_init, dim3(B), dim3(K_), 0, stream,
                     x, prior_w, emw, p0, T);

  const float* src = p0;
  float* dst = out;
  for (int t = 1; t < T; ++t) {
    if (useScaled) {
      hipLaunchKernelGGL(hmm_step<1>, dim3(B), dim3(K_), 0, stream,
                         src, dst, coeff, bias2, emw, x, t, T);
    } else {
      hipLaunchKernelGGL(hmm_step<0>, dim3(B), dim3(K_), 0, stream,
                         src, dst, coeff, bias, emw, x, t, T);
    }
    const float* ns = dst;
    dst = (dst == out) ? bufA : out;
    src = ns;
  }
  if (T > 1 && src != out) {            // T-1 even => result still in bufA
    hipMemcpyAsync(out, (const void*)src, (size_t)B * K * sizeof(float),
                   hipMemcpyDeviceToDevice, stream);
  }
}